// AspectLevelSentimentClassification_21380347199761
// MI455X (gfx1250) — compile-verified
//
#include <hip/hip_runtime.h>

typedef float v2f __attribute__((ext_vector_type(2)));
typedef float v8f __attribute__((ext_vector_type(8)));

#define B_SZ 64
#define S_SZ 4096
#define H_SZ 768
#define C_SZ 3
#define TWOH 1536
#define FPAD 4                 // LDS row pad: stride 1540 -> conflict-free ds_load_b64
#define FROW (TWOH + FPAD)     // 1540 floats per feat row

// One block per 16-batch M-tile: 4 blocks x 256 threads (8 wave32 each).
__global__ __launch_bounds__(256)
void AspectLevelSentimentClassification_kernel(
    const float* __restrict__ lhs,      // (64, 4096, 768)
    const float* __restrict__ pooled,   // (64, 768)
    const int*   __restrict__ posidx,   // (64, 2)
    const float* __restrict__ Wm,       // (3, 1536)
    const float* __restrict__ bias,     // (3,)
    float*       __restrict__ out)      // (64, 3)
{
    __shared__ float s_feat[16][FROW];     // 98560 B: [pooled | aspect] per batch row
    __shared__ float s_partial[8][256];    //  8192 B: per-wave 16x16 D partials

    const int tid = threadIdx.x;
    const int b0  = blockIdx.x * 16;       // first batch of this tile

    // ---------------- Phase 1: build feat tile in LDS ------------------------
    // 16 batches x 768 features = 12288 items, 48 per thread.
    // For fixed j the whole block shares one batch -> uniform row-loop count;
    // lanes cover consecutive h -> coalesced 128B global reads.
    for (int j = 0; j < 48; ++j) {
        const int i     = tid + j * 256;
        const int lb    = i / H_SZ;
        const int h     = i - lb * H_SZ;
        const int batch = b0 + lb;
        // pooled half: straight coalesced copy
        s_feat[lb][h] = pooled[(size_t)batch * H_SZ + h];
        // aspect half: masked mean over rows [start, end)
        const int start = posidx[2 * batch];
        const int end   = posidx[2 * batch + 1];
        const int cnt   = end - start;     // reference guarantees >= 1
        const float* p  = lhs + ((size_t)batch * S_SZ + start) * H_SZ + h;
        float acc = 0.0f;
        for (int r = 0; r < cnt; ++r) acc += p[(size_t)r * H_SZ];
        s_feat[lb][H_SZ + h] = acc / (float)cnt;
    }
    __syncthreads();

    // ---------------- Phase 2: fp32 WMMA GEMV, all 8 waves -------------------
    // D(16x16) += A(16x4) x B(4x16); K = 1536 split 8 ways (48 steps/wave).
    // No divergent control flow: EXEC is all-1s at every WMMA (ISA req).
    const int   wave   = tid >> 5;
    const int   lane   = tid & 31;
    const int   half   = lane >> 4;        // 0: K pair {0,1}, 1: K pair {2,3}
    const int   mrow   = lane & 15;        // A row (M) == B/D column (N)
    const int   kbase  = wave * 192;
    // Branchless B zero-padding: lanes N>=3 load a valid duplicate row, x0.
    const float bscale = (mrow < C_SZ) ? 1.0f : 0.0f;
    const float* wrow  = Wm + (size_t)((mrow < C_SZ) ? mrow : (C_SZ - 1)) * TWOH;

    v8f accs[4] = {v8f{}, v8f{}, v8f{}, v8f{}};   // 4-way ILP: dep chain 48 -> 12
    for (int step = 0; step < 48; step += 4) {
#pragma unroll
        for (int u = 0; u < 4; ++u) {
            const int col = kbase + (step + u) * 4 + 2 * half;  // even -> 8B aligned
            const v2f a  = *(const v2f*)&s_feat[mrow][col];     // ds_load_b64
            v2f       bv = *(const v2f*)&wrow[col];             // global_load_b64
            bv.x *= bscale;
            bv.y *= bscale;
            accs[u] = __builtin_amdgcn_wmma_f32_16x16x4_f32(
                false, a, false, bv, (short)0, accs[u], false, false);
        }
    }
    const v8f acc = (accs[0] + accs[1]) + (accs[2] + accs[3]);

    // D layout: VGPR v, lane -> (M = v + 8*half, N = mrow)
#pragma unroll
    for (int v = 0; v < 8; ++v) {
        s_partial[wave][(v + 8 * half) * 16 + mrow] = acc[v];
    }
    __syncthreads();

    // ---------------- Reduce 8 wave partials, add bias, store ---------------
    {
        const int M = tid >> 4;            // 0..15
        const int N = tid & 15;            // 0..15, only N<3 is real
        if (N < C_SZ) {
            float v = 0.0f;
#pragma unroll
            for (int w = 0; w < 8; ++w) v += s_partial[w][M * 16 + N];
            out[(b0 + M) * C_SZ + N] = v + bias[N];
        }
    }
}

extern "C" void kernel_launch(void* const* d_in, const int* in_sizes, int n_in,
                              void* d_out, int out_size, void* d_ws, size_t ws_size,
                              hipStream_t stream) {
    (void)in_sizes; (void)n_in; (void)out_size; (void)d_ws; (void)ws_size;
    const float* lhs    = (const float*)d_in[0];  // last_hidden_state
    const float* pooled = (const float*)d_in[1];  // pooled_output
    const int*   posidx = (const int*)d_in[2];    // position_indices
    const float* Wm     = (const float*)d_in[3];  // W (3, 1536)
    const float* bias   = (const float*)d_in[4];  // b (3,)
    float*       out    = (float*)d_out;          // (64, 3)

    AspectLevelSentimentClassification_kernel<<<4, 256, 0, stream>>>(
        lhs, pooled, posidx, Wm, bias, out);
}